// BBBEmbedding_12335146074866
// MI455X (gfx1250) — compile-verified
//
#include <hip/hip_runtime.h>
#include <hip/hip_bf16.h>
#include <math.h>

typedef __attribute__((ext_vector_type(4))) float v4f;
typedef __attribute__((ext_vector_type(4))) int   v4i;

#define AS1 __attribute__((address_space(1)))
#define AS3 __attribute__((address_space(3)))

#define DIM             128   // embedding dim (reference: D=128), 32 lanes x float4
#define WAVES_PER_BLOCK 8
#define TOK_PER_WAVE    16
#define TOK_PER_BLOCK   (WAVES_PER_BLOCK * TOK_PER_WAVE)

#if defined(__gfx1250__) &&                                              \
    __has_builtin(__builtin_amdgcn_global_load_async_to_lds_b128) &&     \
    __has_builtin(__builtin_amdgcn_global_store_async_from_lds_b128) &&  \
    __has_builtin(__builtin_amdgcn_s_wait_asynccnt)
#define USE_ASYNC_LDS 1
#else
#define USE_ASYNC_LDS 0
#endif

// ---------------------------------------------------------------------------
// Pass 2: gather.  One wave32 moves one 512B row (float4 per lane) through
// the CDNA5 async data mover: global -> LDS -> global, no VGPR data traffic,
// 16 rows in flight per wave, fenced by s_wait_asynccnt.  Table loads use RT
// policy (row reuse ~8x in the 192MB L2); output stores use NT (write-once
// stream, don't evict the table).
// ---------------------------------------------------------------------------
__global__ __launch_bounds__(256) void gather_kernel(
    const int* __restrict__ x, const float* __restrict__ table,
    float* __restrict__ out, int n_tok) {
  const int wave = threadIdx.x >> 5;
  const int lane = threadIdx.x & 31;
  long base = (long)blockIdx.x * TOK_PER_BLOCK + (long)wave * TOK_PER_WAVE;
  long rem  = (long)n_tok - base;
  if (rem <= 0) return;
  int cnt = rem < TOK_PER_WAVE ? (int)rem : TOK_PER_WAVE;

#if USE_ASYNC_LDS
  // 16B alignment required for the async B128 LDS addresses.
  __shared__ __align__(16) float stage[TOK_PER_BLOCK * DIM];  // 64 KB of 320 KB
  float* wbuf = &stage[wave * TOK_PER_WAVE * DIM];

  // One coalesced vmem op fetches all indices for this wave's batch.
  int myidx = 0;
  if (lane < cnt) myidx = x[base + lane];

  if (cnt == TOK_PER_WAVE) {
    // Fast path: fully unrolled, 16 async row-loads in flight.
#pragma unroll
    for (int t = 0; t < TOK_PER_WAVE; ++t) {
      int idx = __builtin_amdgcn_readlane(myidx, t);
      const float* src = table + (long)idx * DIM + lane * 4;
      __builtin_amdgcn_global_load_async_to_lds_b128(
          (AS1 v4i*)src, (AS3 v4i*)(wbuf + t * DIM + lane * 4),
          /*offset=*/0, /*cpol=*/0);
    }
    __builtin_amdgcn_s_wait_asynccnt(0);         // rows landed in LDS
#pragma unroll
    for (int t = 0; t < TOK_PER_WAVE; ++t) {
      float* dst = out + (base + t) * (long)DIM + lane * 4;
      __builtin_amdgcn_global_store_async_from_lds_b128(
          (AS1 v4i*)dst, (AS3 v4i*)(wbuf + t * DIM + lane * 4),
          /*offset=*/0, /*cpol=*/1);             // TH=NT: streaming output
    }
    __builtin_amdgcn_s_wait_asynccnt(0);
  } else {
    // Generic tail path.
    for (int t = 0; t < cnt; ++t) {
      int idx = __builtin_amdgcn_readlane(myidx, t);
      const float* src = table + (long)idx * DIM + lane * 4;
      __builtin_amdgcn_global_load_async_to_lds_b128(
          (AS1 v4i*)src, (AS3 v4i*)(wbuf + t * DIM + lane * 4), 0, 0);
    }
    __builtin_amdgcn_s_wait_asynccnt(0);
    for (int t = 0; t < cnt; ++t) {
      float* dst = out + (base + t) * (long)DIM + lane * 4;
      __builtin_amdgcn_global_store_async_from_lds_b128(
          (AS1 v4i*)dst, (AS3 v4i*)(wbuf + t * DIM + lane * 4), 0, 1);
    }
    __builtin_amdgcn_s_wait_asynccnt(0);
  }
#else
  // Fallback: VGPR copy path (still coalesced float4 per lane).
  for (int t = 0; t < cnt; ++t) {
    int idx = __builtin_amdgcn_readfirstlane(x[base + t]);
    const v4f* src = (const v4f*)(table + (long)idx * DIM) + lane;
    v4f* dst = (v4f*)(out + (base + t) * (long)DIM) + lane;
    __builtin_nontemporal_store(*src, dst);
  }
#endif
}

// ---------------------------------------------------------------------------
// Pass 1: sample the whole table once.  table[v,d] = mu + softplus(rho)*clip(eps)
// Sources are read-once -> non-temporal loads (don't pollute L2).
// The 51.2 MB result table is written with default (RT) policy so it stays
// resident in the 192 MB L2 for the gather pass.
// ---------------------------------------------------------------------------
__global__ __launch_bounds__(256) void sample_table_kernel(
    const float* __restrict__ mu, const float* __restrict__ rho,
    const float* __restrict__ eps, float* __restrict__ table, int n4) {
  int i = blockIdx.x * blockDim.x + threadIdx.x;
  if (i >= n4) return;
  const v4f* m4 = (const v4f*)mu;
  const v4f* r4 = (const v4f*)rho;
  const v4f* e4 = (const v4f*)eps;
  v4f m = __builtin_nontemporal_load(m4 + i);
  v4f r = __builtin_nontemporal_load(r4 + i);
  v4f e = __builtin_nontemporal_load(e4 + i);
  v4f o;
#pragma unroll
  for (int c = 0; c < 4; ++c) {
    float ec = fminf(fmaxf(e[c], -10.0f), 10.0f);
    float s  = log1pf(expf(r[c]));        // softplus, matches jnp.log1p(jnp.exp())
    o[c]     = fmaf(s, ec, m[c]);
  }
  ((v4f*)table)[i] = o;
}

// ---------------------------------------------------------------------------
// Fallback if workspace cannot hold the sampled table: fused gather+sample.
// One wave per token; lane handles a float4 chunk of the row.
// ---------------------------------------------------------------------------
__global__ __launch_bounds__(256) void fused_gather_kernel(
    const int* __restrict__ x, const float* __restrict__ mu,
    const float* __restrict__ rho, const float* __restrict__ eps,
    float* __restrict__ out, int n_tok) {
  long gtid = (long)blockIdx.x * blockDim.x + threadIdx.x;
  long tok  = gtid >> 5;          // 32 lanes per token (wave32)
  int lane  = (int)(gtid & 31);
  if (tok >= n_tok) return;
  int idx = __builtin_amdgcn_readfirstlane(x[tok]);
  long off = (long)idx * DIM + lane * 4;
  v4f m = *(const v4f*)(mu + off);
  v4f r = *(const v4f*)(rho + off);
  v4f e = *(const v4f*)(eps + off);
  v4f o;
#pragma unroll
  for (int c = 0; c < 4; ++c) {
    float ec = fminf(fmaxf(e[c], -10.0f), 10.0f);
    float s  = log1pf(expf(r[c]));
    o[c]     = fmaf(s, ec, m[c]);
  }
  __builtin_nontemporal_store(o, (v4f*)(out + tok * (long)DIM + lane * 4));
}

extern "C" void kernel_launch(void* const* d_in, const int* in_sizes, int n_in,
                              void* d_out, int out_size, void* d_ws, size_t ws_size,
                              hipStream_t stream) {
  const int*   x   = (const int*)d_in[0];
  const float* mu  = (const float*)d_in[1];
  const float* rho = (const float*)d_in[2];
  const float* eps = (const float*)d_in[3];
  float* out = (float*)d_out;

  const long n_tok = in_sizes[0];      // B*L
  const long vd    = in_sizes[1];      // V*D
  const size_t need = (size_t)vd * sizeof(float);

  if (ws_size >= need) {
    float* table = (float*)d_ws;
    int n4 = (int)(vd / 4);
    int blk1 = (n4 + 255) / 256;
    sample_table_kernel<<<blk1, 256, 0, stream>>>(mu, rho, eps, table, n4);
    long blk2 = (n_tok + TOK_PER_BLOCK - 1) / TOK_PER_BLOCK;
    gather_kernel<<<(int)blk2, 256, 0, stream>>>(x, table, out, (int)n_tok);
  } else {
    long threads = n_tok * 32;
    long blk = (threads + 255) / 256;
    fused_gather_kernel<<<(int)blk, 256, 0, stream>>>(x, mu, rho, eps, out,
                                                      (int)n_tok);
  }
}